// RecursiveUpdateLayer_46591805227680
// MI455X (gfx1250) — compile-verified
//
#include <hip/hip_runtime.h>
#include <math.h>

typedef __attribute__((ext_vector_type(2))) float v2f;
typedef __attribute__((ext_vector_type(8))) float v8f;

#define B_    8
#define C_    256
#define H_    64
#define W_    64
#define HW_   4096
#define J_    17
#define JD_   51      /* J*dim */
#define MPAD_ 320     /* padded combined 1x1 output channels (289 -> 320) */

/* workspace layout (float offsets). s_all aliases off27+dcn_out (dead after K4). */
#define WS_OFF27  ((size_t)0)          /* 8*27*4096   =   884736 */
#define WS_DCN    ((size_t)884736)     /* 8*256*4096  =  8388608 */
#define WS_SALL   ((size_t)0)          /* 1088*2*4096 =  8912896 (aliases above) */
#define WS_WK     ((size_t)9273344)    /* 9*256*256   =   589824 */
#define WS_WCAT   ((size_t)9863168)    /* 320*256     =    81920 */
#define WS_BCAT   ((size_t)9945088)    /* 320 */
#define WS_GNST   ((size_t)9945408)    /* 8*32*2      =      512 */
#define WS_CONV   ((size_t)9945920)    /* 8*320*4096  = 10485760 */
#define WS_OFFS   ((size_t)20431680)   /* 8*51*4096   =  1671168 */
#define WS_WOFF   ((size_t)22102848)   /* 9*32*256    =    73728 */

__device__ __forceinline__ v8f wmma4(v2f a, v2f b, v8f c) {
  return __builtin_amdgcn_wmma_f32_16x16x4_f32(false, a, false, b, (short)0, c, false, false);
}

struct Corners { float w0, w1, w2, w3; int i0, i1, i2, i3; };

__device__ __forceinline__ Corners mk_corners(float sx, float sy) {
  float x0f = floorf(sx), y0f = floorf(sy);
  int x0 = (int)x0f, y0 = (int)y0f;
  float fx = sx - x0f, fy = sy - y0f;
  float vx0 = (x0 >= 0 && x0 < W_)       ? 1.f : 0.f;
  float vx1 = (x0 + 1 >= 0 && x0 + 1 < W_) ? 1.f : 0.f;
  float vy0 = (y0 >= 0 && y0 < H_)       ? 1.f : 0.f;
  float vy1 = (y0 + 1 >= 0 && y0 + 1 < H_) ? 1.f : 0.f;
  int cx0 = min(max(x0, 0), W_ - 1), cx1 = min(max(x0 + 1, 0), W_ - 1);
  int cy0 = min(max(y0, 0), H_ - 1), cy1 = min(max(y0 + 1, 0), H_ - 1);
  Corners c;
  c.w0 = (1.f - fx) * (1.f - fy) * vx0 * vy0;
  c.w1 = fx * (1.f - fy) * vx1 * vy0;
  c.w2 = (1.f - fx) * fy * vx0 * vy1;
  c.w3 = fx * fy * vx1 * vy1;
  c.i0 = cy0 * W_ + cx0; c.i1 = cy0 * W_ + cx1;
  c.i2 = cy1 * W_ + cx0; c.i3 = cy1 * W_ + cx1;
  return c;
}

__device__ __forceinline__ float csample(const float* __restrict__ p, const Corners& c) {
  return p[c.i0] * c.w0 + p[c.i1] * c.w1 + p[c.i2] * c.w2 + p[c.i3] * c.w3;
}

__device__ __forceinline__ float sigm(float x) { return 1.f / (1.f + __expf(-x)); }

/* ---- K0a: pack DCN weights: wk[k][oc][ic] = dcn_w[oc][ic][k/3][k%3] ---- */
__global__ __launch_bounds__(256) void k_pack_wk(const float* __restrict__ dcn_w,
                                                 float* __restrict__ wk) {
  int i = blockIdx.x * 256 + threadIdx.x;           /* 9*256*256 = 589824 */
  if (i >= 9 * C_ * C_) return;
  int ic = i & 255, oc = (i >> 8) & 255, k = i >> 16;
  wk[i] = dcn_w[(size_t)(oc * C_ + ic) * 9 + k];
}

/* ---- K0b: pack offset-conv weights: wofft[t][o(32 pad)][ic] ---- */
__global__ __launch_bounds__(256) void k_pack_woff(const float* __restrict__ dcn_off_w,
                                                   float* __restrict__ wofft) {
  int i = blockIdx.x * 256 + threadIdx.x;           /* 9*32*256 = 73728 */
  if (i >= 9 * 32 * C_) return;
  int ic = i & 255, o = (i >> 8) & 31, t = i >> 13;
  wofft[i] = (o < 27) ? dcn_off_w[(size_t)(o * C_ + ic) * 9 + t] : 0.f;
}

/* ---- K0c: pack concatenated 1x1 weights [320][256] + bias[320] ---- */
__global__ __launch_bounds__(256) void k_pack_wcat(
    const float* __restrict__ so_w, const float* __restrict__ so_b,
    const float* __restrict__ sc_w, const float* __restrict__ sc_b,
    const float* __restrict__ uw_w, const float* __restrict__ uw_b,
    const float* __restrict__ uo_w, const float* __restrict__ uo_b,
    float* __restrict__ wcat, float* __restrict__ bcat) {
  int i = blockIdx.x * 256 + threadIdx.x;           /* 320*256 = 81920 */
  if (i >= MPAD_ * C_) return;
  int c = i & 255, r = i >> 8;
  float w = 0.f, b = 0.f;
  if (r < 136)      { w = so_w[r * C_ + c];        b = so_b[r]; }
  else if (r < 187) { w = sc_w[(r - 136) * C_ + c]; b = sc_b[r - 136]; }
  else if (r < 238) { w = uw_w[(r - 187) * C_ + c]; b = uw_b[r - 187]; }
  else if (r < 289) { w = uo_w[(r - 238) * C_ + c]; b = uo_b[r - 238]; }
  wcat[i] = w;
  if (c == 0) bcat[r] = b;
}

/* ---- K1: 3x3 offset conv as 9 shifted 1x1 WMMA GEMMs (27 ch, pad 32) ---- */
__global__ __launch_bounds__(256) void k_offconv_wmma(const float* __restrict__ feat,
                                                      const float* __restrict__ wofft,
                                                      const float* __restrict__ boff,
                                                      float* __restrict__ off27) {
  __shared__ float Smat[C_][64];                    /* 64 KB shifted tile */
  const int tid  = threadIdx.x;
  const int b    = blockIdx.y;
  const int y    = blockIdx.x;                      /* image row */
  const int p0   = y * 64;
  const int wave = tid >> 5, lane = tid & 31;
  const int lrow = lane & 15, lhi = lane >> 4;
  const int px = tid & 63, cg = tid >> 6;
  const int m = wave & 1;                           /* M-tile 0..1 */
  const int n = wave >> 1;                          /* N-tile 0..3 */

  v8f acc0 = {}, acc1 = {};                         /* two K-chains */

  for (int t = 0; t < 9; ++t) {
    __syncthreads();
    /* stage shifted feature tile (zero borders) */
    const int yy = y + (t / 3) - 1;
    const int xx = px + (t % 3) - 1;
    const bool valid = (yy >= 0 && yy < H_ && xx >= 0 && xx < W_);
    const float* fb = feat + ((size_t)b * C_ + cg * 64) * HW_;
    const int src = valid ? (yy * W_ + xx) : 0;
    const float vm = valid ? 1.f : 0.f;
    for (int cc = 0; cc < 64; ++cc)
      Smat[cg * 64 + cc][px] = fb[(size_t)cc * HW_ + src] * vm;
    __syncthreads();
    /* GEMM: acc += Woff_t(32x256) * S(256x64), two interleaved chains */
    const float* wt = wofft + (size_t)t * (32 * C_);
    const float* wrow = wt + (size_t)((m * 16 + lrow) * C_) + 2 * lhi;
    for (int kk = 0; kk < C_; kk += 8) {
      v2f a0 = *(const v2f*)(wrow + kk);
      v2f a1 = *(const v2f*)(wrow + kk + 4);
      v2f b0, b1;
      b0.x = Smat[kk + 2 * lhi][n * 16 + lrow];
      b0.y = Smat[kk + 1 + 2 * lhi][n * 16 + lrow];
      b1.x = Smat[kk + 4 + 2 * lhi][n * 16 + lrow];
      b1.y = Smat[kk + 5 + 2 * lhi][n * 16 + lrow];
      acc0 = wmma4(a0, b0, acc0);
      acc1 = wmma4(a1, b1, acc1);
    }
  }
  v8f accs = acc0 + acc1;
  #pragma unroll
  for (int r = 0; r < 8; ++r) {
    int oc = m * 16 + r + 8 * lhi;
    if (oc < 27) {
      float v = accs[r] + boff[oc];
      if (oc >= 18) v = sigm(v);
      off27[((size_t)(b * 27 + oc)) * HW_ + p0 + n * 16 + lrow] = v;
    }
  }
}

/* ---- K2: DCN — gather deformed tile to LDS, 9-tap f32 WMMA GEMM ---- */
__global__ __launch_bounds__(256) void k_dcn_wmma(const float* __restrict__ feat,
                                                  const float* __restrict__ off27,
                                                  const float* __restrict__ wk,
                                                  float* __restrict__ dcn_out) {
  __shared__ float Smat[C_][64];                    /* 64 KB: B-matrix tile */
  const int tid  = threadIdx.x;
  const int b    = blockIdx.y;
  const int tile = blockIdx.x;                      /* one image row of 64 px */
  const int p0   = tile * 64;
  const int wave = tid >> 5, lane = tid & 31;
  const int lrow = lane & 15, lhi = lane >> 4;      /* WMMA lane coords */
  const int px = tid & 63, cg = tid >> 6;           /* sampling role */
  const float xf = (float)px, yf = (float)tile;

  v8f acc[2][4] = {};                               /* 2 M-tiles x 4 N-tiles */

  for (int k = 0; k < 9; ++k) {
    __syncthreads();
    /* gather: 4 threads/pixel, 64 channels each, mask folded into weights */
    const int pp = p0 + px;
    float dy = off27[(size_t)(b * 27 + k) * HW_ + pp];
    float dx = off27[(size_t)(b * 27 + 9 + k) * HW_ + pp];
    float mk = off27[(size_t)(b * 27 + 18 + k) * HW_ + pp];
    float sy = yf + (float)(k / 3 - 1) + dy;
    float sx = xf + (float)(k % 3 - 1) + dx;
    Corners c = mk_corners(sx, sy);
    c.w0 *= mk; c.w1 *= mk; c.w2 *= mk; c.w3 *= mk;
    const float* fb = feat + ((size_t)b * C_ + cg * 64) * HW_;
    for (int cc = 0; cc < 64; ++cc)
      Smat[cg * 64 + cc][px] = csample(fb + (size_t)cc * HW_, c);
    __syncthreads();
    /* GEMM: D += Wk(256x256) * S(256x64) */
    const float* wkk = wk + (size_t)k * (C_ * C_);
    const int m0 = wave * 2;
    for (int kk = 0; kk < C_; kk += 4) {
      v2f a0 = *(const v2f*)(wkk + (size_t)((m0 * 16 + lrow) * C_) + kk + 2 * lhi);
      v2f a1 = *(const v2f*)(wkk + (size_t)(((m0 + 1) * 16 + lrow) * C_) + kk + 2 * lhi);
      #pragma unroll
      for (int n = 0; n < 4; ++n) {
        v2f bb;
        bb.x = Smat[kk + 2 * lhi][n * 16 + lrow];
        bb.y = Smat[kk + 1 + 2 * lhi][n * 16 + lrow];
        acc[0][n] = wmma4(a0, bb, acc[0][n]);
        acc[1][n] = wmma4(a1, bb, acc[1][n]);
      }
    }
  }
  #pragma unroll
  for (int mi = 0; mi < 2; ++mi) {
    const int ocb = (wave * 2 + mi) * 16;
    #pragma unroll
    for (int n = 0; n < 4; ++n)
      #pragma unroll
      for (int r = 0; r < 8; ++r) {
        int oc = ocb + r + 8 * lhi;
        dcn_out[((size_t)(b * C_ + oc)) * HW_ + p0 + n * 16 + lrow] = acc[mi][n][r];
      }
  }
}

/* ---- K3: GroupNorm stats (32 groups of 8 channels) ---- */
__global__ __launch_bounds__(256) void k_gn_stats(const float* __restrict__ dcn,
                                                  float* __restrict__ st) {
  __shared__ float ss[256], qq[256];
  int g = blockIdx.x, b = blockIdx.y, t = threadIdx.x;
  const float* base = dcn + ((size_t)(b * C_ + g * 8)) * HW_;
  float s = 0.f, q = 0.f;
  for (int i = t; i < 8 * HW_; i += 256) { float v = base[i]; s += v; q += v * v; }
  ss[t] = s; qq[t] = q; __syncthreads();
  for (int k = 128; k > 0; k >>= 1) {
    if (t < k) { ss[t] += ss[t + k]; qq[t] += qq[t + k]; }
    __syncthreads();
  }
  if (t == 0) {
    float inv = 1.f / (8.f * HW_);
    float mu = ss[0] * inv;
    st[(b * 32 + g) * 2] = mu;
    st[(b * 32 + g) * 2 + 1] = qq[0] * inv - mu * mu;
  }
}

/* ---- K4: feat2 = feat + relu(GN(dcn_out)) -> d_out feat region ---- */
__global__ __launch_bounds__(256) void k_gn_apply(const float* __restrict__ dcn,
                                                  const float* __restrict__ st,
                                                  const float* __restrict__ gn_g,
                                                  const float* __restrict__ gn_b,
                                                  const float* __restrict__ feat,
                                                  float* __restrict__ out_feat) {
  int i = blockIdx.x * 256 + threadIdx.x;           /* 8*256*4096 */
  if (i >= B_ * C_ * HW_) return;
  int c = (i >> 12) & 255, b = i >> 20;
  float mu  = st[(b * 32 + c / 8) * 2];
  float var = st[(b * 32 + c / 8) * 2 + 1];
  float xn = (dcn[i] - mu) * rsqrtf(var + 1e-5f);
  float v = xn * gn_g[c] + gn_b[c];
  out_feat[i] = feat[i] + fmaxf(v, 0.f);
}

/* ---- K5: fused four 1x1 convs as one 320x256 WMMA GEMM ---- */
__global__ __launch_bounds__(256) void k_conv1x1_wmma(const float* __restrict__ feat2,
                                                      const float* __restrict__ wcat,
                                                      const float* __restrict__ bcat,
                                                      float* __restrict__ conv_out) {
  __shared__ float Fmat[C_][64];                    /* 64 KB */
  const int tid = threadIdx.x;
  const int b = blockIdx.z, mchunk = blockIdx.y, tile = blockIdx.x;
  const int p0 = tile * 64;
  const int wave = tid >> 5, lane = tid & 31;
  const int lrow = lane & 15, lhi = lane >> 4;
  const int px = tid & 63, cg = tid >> 6;

  const float* fb = feat2 + ((size_t)b * C_ + cg * 64) * HW_ + p0;
  for (int cc = 0; cc < 64; ++cc)
    Fmat[cg * 64 + cc][px] = fb[(size_t)cc * HW_ + px];
  __syncthreads();

  const int m = mchunk * 4 + (wave & 3);
  const int nb = 2 * (wave >> 2);
  v8f acc[2] = {};
  for (int kk = 0; kk < C_; kk += 4) {
    v2f a = *(const v2f*)(wcat + (size_t)((m * 16 + lrow) * C_) + kk + 2 * lhi);
    #pragma unroll
    for (int ni = 0; ni < 2; ++ni) {
      int n = nb + ni;
      v2f bb;
      bb.x = Fmat[kk + 2 * lhi][n * 16 + lrow];
      bb.y = Fmat[kk + 1 + 2 * lhi][n * 16 + lrow];
      acc[ni] = wmma4(a, bb, acc[ni]);
    }
  }
  #pragma unroll
  for (int ni = 0; ni < 2; ++ni) {
    int n = nb + ni;
    #pragma unroll
    for (int r = 0; r < 8; ++r) {
      int oc = m * 16 + r + 8 * lhi;
      conv_out[((size_t)(b * MPAD_ + oc)) * HW_ + p0 + n * 16 + lrow] = acc[ni][r] + bcat[oc];
    }
  }
}

/* ---- K6: offset = (1-sig(ow))*prev + sig(ow)*uo ---- */
__global__ __launch_bounds__(256) void k_update_offset(const float* __restrict__ conv_out,
                                                       const float* __restrict__ prev,
                                                       float* __restrict__ offs) {
  int i = blockIdx.x * 256 + threadIdx.x;           /* 8*51*4096 */
  if (i >= B_ * JD_ * HW_) return;
  int p = i & 4095, c = (i >> 12) % JD_, b = i / (JD_ * HW_);
  float s  = sigm(conv_out[((size_t)(b * MPAD_ + 187 + c)) * HW_ + p]);
  float uo = conv_out[((size_t)(b * MPAD_ + 238 + c)) * HW_ + p];
  offs[i] = (1.f - s) * prev[i] + s * uo;
}

/* ---- K7: build s_all = concat(grid_sample(soff)+off, soff) ---- */
__global__ __launch_bounds__(256) void k_sall(const float* __restrict__ offs,
                                              const float* __restrict__ conv_out,
                                              float* __restrict__ s_all) {
  int i = blockIdx.x * 256 + threadIdx.x;           /* 8*17*4096 */
  if (i >= B_ * J_ * HW_) return;
  int p = i & 4095, j = (i >> 12) % J_, b = i / (J_ * HW_);
  float u = offs[((size_t)(b * JD_ + j * 3)) * HW_ + p];
  float v = offs[((size_t)(b * JD_ + j * 3 + 1)) * HW_ + p];
  float pxf = (float)(p & 63), pyf = (float)(p >> 6);
  Corners c = mk_corners(pxf + u, pyf + v);
  float* sab = s_all + ((size_t)((b * J_ + j) * 8)) * 2 * HW_;
  #pragma unroll
  for (int hd = 0; hd < 4; ++hd) {
    const float* pl = conv_out + ((size_t)(b * MPAD_ + j * 8 + hd * 2)) * HW_;
    sab[(size_t)(hd * 2 + 0) * HW_ + p] = csample(pl, c) + u;
    sab[(size_t)(hd * 2 + 1) * HW_ + p] = csample(pl + HW_, c) + v;
    sab[(size_t)((4 + hd) * 2 + 0) * HW_ + p] = pl[p];
    sab[(size_t)((4 + hd) * 2 + 1) * HW_ + p] = pl[HW_ + p];
  }
}

/* ---- K8: second sample (u,v,d,conf), softmax over 8 heads, reduce ---- */
__global__ __launch_bounds__(256) void k_final(const float* __restrict__ offs,
                                               const float* __restrict__ conv_out,
                                               const float* __restrict__ s_all,
                                               float* __restrict__ out_off) {
  int i = blockIdx.x * 256 + threadIdx.x;           /* 8*17*4096 */
  if (i >= B_ * J_ * HW_) return;
  int p = i & 4095, j = (i >> 12) % J_, b = i / (J_ * HW_);
  const float* up  = offs + ((size_t)(b * JD_ + j * 3)) * HW_;
  const float* cp  = conv_out + ((size_t)(b * MPAD_ + 136 + j * 3)) * HW_;
  const float* sab = s_all + ((size_t)((b * J_ + j) * 8)) * 2 * HW_;
  float pxf = (float)(p & 63), pyf = (float)(p >> 6);
  float soh[8][3], sch[8][3];
  #pragma unroll
  for (int h = 0; h < 8; ++h) {
    float sx = sab[(size_t)(h * 2 + 0) * HW_ + p];
    float sy = sab[(size_t)(h * 2 + 1) * HW_ + p];
    Corners c = mk_corners(pxf + sx, pyf + sy);
    soh[h][0] = csample(up, c) + sx;
    soh[h][1] = csample(up + HW_, c) + sy;
    soh[h][2] = csample(up + 2 * HW_, c);
    sch[h][0] = csample(cp, c);
    sch[h][1] = csample(cp + HW_, c);
    sch[h][2] = csample(cp + 2 * HW_, c);
  }
  #pragma unroll
  for (int di = 0; di < 3; ++di) {
    float mx = sch[0][di];
    #pragma unroll
    for (int h = 1; h < 8; ++h) mx = fmaxf(mx, sch[h][di]);
    float se = 0.f, ac = 0.f;
    #pragma unroll
    for (int h = 0; h < 8; ++h) {
      float e = __expf(sch[h][di] - mx);
      se += e;
      ac += soh[h][di] * e;
    }
    out_off[((size_t)(b * JD_ + j * 3 + di)) * HW_ + p] = ac / se;
  }
}

extern "C" void kernel_launch(void* const* d_in, const int* in_sizes, int n_in,
                              void* d_out, int out_size, void* d_ws, size_t ws_size,
                              hipStream_t stream) {
  const float* feat      = (const float*)d_in[0];
  const float* prev      = (const float*)d_in[1];
  const float* dcn_off_w = (const float*)d_in[2];
  const float* dcn_off_b = (const float*)d_in[3];
  const float* dcn_w     = (const float*)d_in[4];
  const float* gn_g      = (const float*)d_in[5];
  const float* gn_b      = (const float*)d_in[6];
  const float* so_w      = (const float*)d_in[7];
  const float* so_b      = (const float*)d_in[8];
  const float* sc_w      = (const float*)d_in[9];
  const float* sc_b      = (const float*)d_in[10];
  const float* uw_w      = (const float*)d_in[11];
  const float* uw_b      = (const float*)d_in[12];
  const float* uo_w      = (const float*)d_in[13];
  const float* uo_b      = (const float*)d_in[14];

  float* ws = (float*)d_ws;
  float* out_feat = (float*)d_out;
  float* out_off  = out_feat + (size_t)B_ * C_ * HW_;

  k_pack_wk<<<2304, 256, 0, stream>>>(dcn_w, ws + WS_WK);
  k_pack_woff<<<288, 256, 0, stream>>>(dcn_off_w, ws + WS_WOFF);
  k_pack_wcat<<<320, 256, 0, stream>>>(so_w, so_b, sc_w, sc_b, uw_w, uw_b, uo_w, uo_b,
                                       ws + WS_WCAT, ws + WS_BCAT);
  k_offconv_wmma<<<dim3(64, 8), 256, 0, stream>>>(feat, ws + WS_WOFF, dcn_off_b,
                                                  ws + WS_OFF27);
  k_dcn_wmma<<<dim3(64, 8), 256, 0, stream>>>(feat, ws + WS_OFF27, ws + WS_WK, ws + WS_DCN);
  k_gn_stats<<<dim3(32, 8), 256, 0, stream>>>(ws + WS_DCN, ws + WS_GNST);
  k_gn_apply<<<32768, 256, 0, stream>>>(ws + WS_DCN, ws + WS_GNST, gn_g, gn_b, feat, out_feat);
  k_conv1x1_wmma<<<dim3(64, 5, 8), 256, 0, stream>>>(out_feat, ws + WS_WCAT, ws + WS_BCAT,
                                                     ws + WS_CONV);
  k_update_offset<<<6528, 256, 0, stream>>>(ws + WS_CONV, prev, ws + WS_OFFS);
  k_sall<<<2176, 256, 0, stream>>>(ws + WS_OFFS, ws + WS_CONV, ws + WS_SALL);
  k_final<<<2176, 256, 0, stream>>>(ws + WS_OFFS, ws + WS_CONV, ws + WS_SALL, out_off);
}